// CrfRnnLayerSPIO_3942779978352
// MI455X (gfx1250) — compile-verified
//
#include <hip/hip_runtime.h>
#include <hip/hip_bf16.h>

#define NC    6
#define HH    768
#define WW    768
#define PP    (HH * WW)          // 589824
#define SEGN  500
#define NITER 5
#define EPSF  1e-5f

typedef float v2f __attribute__((ext_vector_type(2)));
typedef float v8f __attribute__((ext_vector_type(8)));

// ---------- shared per-pixel math: softmax + "a" term --------------------
__device__ __forceinline__ void crf_sm_a(const float q[NC], float sm[NC], float a[NC]) {
    float m = q[0];
#pragma unroll
    for (int c = 1; c < NC; ++c) m = fmaxf(m, q[c]);
    float e[NC], s = 0.f;
#pragma unroll
    for (int c = 0; c < NC; ++c) { e[c] = __expf(q[c] - m); s += e[c]; }
    float inv = 1.0f / s;
#pragma unroll
    for (int c = 0; c < NC; ++c) sm[c] = e[c] * inv;
    float mx = sm[0];
#pragma unroll
    for (int c = 1; c < NC; ++c) mx = fmaxf(mx, sm[c]);
#pragma unroll
    for (int c = 0; c < NC; ++c) {
        float sub = (sm[c] == mx) ? sm[c] : 0.f;   // bool_max semantics
        a[c] = (sm[c] + EPSF) + (mx + EPSF) - sub; // q_eps + max_q - where(bool_max, sm, 0)
    }
}

__device__ __forceinline__ void load6(const float* __restrict__ p, float q[NC]) {
    const float2* v = reinterpret_cast<const float2*>(p);   // 8B aligned (p*24)
    float2 x = v[0], y = v[1], z = v[2];
    q[0] = x.x; q[1] = x.y; q[2] = y.x; q[3] = y.y; q[4] = z.x; q[5] = z.y;
}

// ---------- one-time: transpose sp_map (seg uses transposed map) ---------
__global__ void k_seg_transpose(const int* __restrict__ src, int* __restrict__ dst) {
    __shared__ int tile[32][33];
    int x = blockIdx.x * 32 + threadIdx.x;
    int y = blockIdx.y * 32 + threadIdx.y;
    tile[threadIdx.y][threadIdx.x] = src[y * WW + x];
    __syncthreads();
    int tx = blockIdx.y * 32 + threadIdx.x;
    int ty = blockIdx.x * 32 + threadIdx.y;
    dst[ty * WW + tx] = tile[threadIdx.x][threadIdx.y];
}

// ---------- one-time: Mtot = compat @ (spatial_w + bilateral_w) ----------
__global__ void k_mtot(const float* __restrict__ compat, const float* __restrict__ sw,
                       const float* __restrict__ bw, float* __restrict__ Mt) {
    int t = threadIdx.x;
    if (t < NC * NC) {
        int i = t / NC, j = t % NC;
        float s = 0.f;
#pragma unroll
        for (int k = 0; k < NC; ++k) s += compat[i * NC + k] * (sw[k * NC + j] + bw[k * NC + j]);
        Mt[t] = s;
    }
}

__global__ void k_zero(float* __restrict__ p, int n) {
    int i = blockIdx.x * blockDim.x + threadIdx.x;
    if (i < n) p[i] = 0.f;
}

// ---------- phase A: segment sums of log(sm+eps) and log(a) --------------
#define PIX_PER_BLK 4096
__global__ void k_accum(const float* __restrict__ qsrc, const int* __restrict__ segt,
                        float* __restrict__ Bsp, float* __restrict__ Bco) {
    __shared__ float lsp[SEGN * NC];
    __shared__ float lco[SEGN * NC];
    for (int i = threadIdx.x; i < SEGN * NC; i += blockDim.x) { lsp[i] = 0.f; lco[i] = 0.f; }
    __syncthreads();

    int base = blockIdx.x * PIX_PER_BLK;
    for (int off = threadIdx.x; off < PIX_PER_BLK; off += blockDim.x) {
        int p = base + off;
        float q[NC], sm[NC], a[NC];
        load6(qsrc + (size_t)p * NC, q);
        crf_sm_a(q, sm, a);
        int sv = segt[p];
        if (sv >= 0 && sv < SEGN) {              // in_r mask
            int b = sv * NC;
#pragma unroll
            for (int c = 0; c < NC; ++c) atomicAdd(&lsp[b + c], __logf(sm[c] + EPSF));
#pragma unroll
            for (int c = 0; c < NC; ++c) atomicAdd(&lco[b + c], __logf(a[c]));
        }
    }
    __syncthreads();
    for (int i = threadIdx.x; i < SEGN * NC; i += blockDim.x) {
        atomicAdd(&Bsp[i], lsp[i]);
        atomicAdd(&Bco[i], lco[i]);
    }
}

// ---------- phase B: gates + WMMA pairwise + q update --------------------
__global__ void k_update(const float* __restrict__ qsrc, const float* __restrict__ un,
                         const int* __restrict__ segt, const float* __restrict__ Bsp,
                         const float* __restrict__ Bco, const float* __restrict__ Mt,
                         const float* __restrict__ lw, const float* __restrict__ hw,
                         float* __restrict__ qdst) {
    int p = blockIdx.x * blockDim.x + threadIdx.x;   // grid exactly covers PP
    float q[NC], sm[NC], a[NC];
    load6(qsrc + (size_t)p * NC, q);
    crf_sm_a(q, sm, a);

    int sv   = segt[p];
    float in_r = (sv >= 0 && sv < SEGN) ? 1.f : 0.f;
    int sc = sv < 0 ? 0 : (sv > SEGN - 1 ? SEGN - 1 : sv);

    float h0 = hw[0], h1 = hw[1];
    float cont[NC];
#pragma unroll
    for (int c = 0; c < NC; ++c) {
        float prod_sp = Bsp[sc * NC + c] * in_r;
        float prod_io = Bco[sc * NC + c] * in_r;
        float qmod  = sm[c] + ((sm[c] == 0.f) ? 1.f : 0.f);
        float ft_sp = __expf(prod_sp - __logf(qmod + EPSF));
        float qvs   = a[c] * in_r;
        qvs += (qvs == 0.f) ? 1.f : 0.f;
        float ft_co = __expf(prod_io - __logf(qvs + EPSF));
        float spu = lw[c]      * ft_sp + h0 * (1.f - ft_sp);
        float cou = lw[NC + c] * ft_co + h1 * (1.f - ft_co);
        cont[c] = spu + cou;
    }

    // pairwise[c] = sum_k Mt[c][k] * sm[k]   (6x6 padded into 16x16x4 WMMA, K=6=4+2)
    float pw[NC];
#if __has_builtin(__builtin_amdgcn_wmma_f32_16x16x4_f32)
    {
        unsigned l = threadIdx.x & 31u;
        int  row = (int)(l & 15u);
        bool lo  = l < 16u;
        int  rr  = row < 6 ? row : 0;
        float rm = row < 6 ? 1.f : 0.f;
        float m0 = Mt[rr * NC + 0] * rm, m1 = Mt[rr * NC + 1] * rm, m2 = Mt[rr * NC + 2] * rm;
        float m3 = Mt[rr * NC + 3] * rm, m4 = Mt[rr * NC + 4] * rm, m5 = Mt[rr * NC + 5] * rm;
        v2f A0, A1;
        A0.x = lo ? m0 : m2;  A0.y = lo ? m1 : m3;   // K-chunk {0,1 | 2,3}
        A1.x = lo ? m4 : 0.f; A1.y = lo ? m5 : 0.f;  // K-chunk {4,5 | pad}

        v8f acc[2];
#pragma unroll
        for (int h = 0; h < 2; ++h) {                // half h owns pixels of lanes h*16..h*16+15
            int src = row + h * 16;
            float s0 = __shfl(sm[0], src), s1 = __shfl(sm[1], src), s2 = __shfl(sm[2], src);
            float s3 = __shfl(sm[3], src), s4 = __shfl(sm[4], src), s5 = __shfl(sm[5], src);
            v2f B0, B1;
            B0.x = lo ? s0 : s2;  B0.y = lo ? s1 : s3;
            B1.x = lo ? s4 : 0.f; B1.y = lo ? s5 : 0.f;
            v8f c0 = {0.f, 0.f, 0.f, 0.f, 0.f, 0.f, 0.f, 0.f};
            c0 = __builtin_amdgcn_wmma_f32_16x16x4_f32(false, A0, false, B0, (short)0, c0, false, false);
            c0 = __builtin_amdgcn_wmma_f32_16x16x4_f32(false, A1, false, B1, (short)0, c0, false, false);
            acc[h] = c0;
        }
#pragma unroll
        for (int c = 0; c < NC; ++c) {               // D: lane n<16, VGPR c = (M=c, N=n)
            float tA = __shfl(acc[0][c], row);
            float tB = __shfl(acc[1][c], row);
            pw[c] = lo ? tA : tB;
        }
    }
#else
#pragma unroll
    for (int c = 0; c < NC; ++c) {
        float s = 0.f;
#pragma unroll
        for (int k = 0; k < NC; ++k) s += Mt[c * NC + k] * sm[k];
        pw[c] = s;
    }
#endif

    const float* up = un + (size_t)p * NC;
    float o[NC];
#pragma unroll
    for (int c = 0; c < NC; ++c) o[c] = up[c] - pw[c] - cont[c];
    float2* out2 = reinterpret_cast<float2*>(qdst + (size_t)p * NC);
    out2[0] = make_float2(o[0], o[1]);
    out2[1] = make_float2(o[2], o[3]);
    out2[2] = make_float2(o[4], o[5]);
}

// -------------------------------------------------------------------------
extern "C" void kernel_launch(void* const* d_in, const int* in_sizes, int n_in,
                              void* d_out, int out_size, void* d_ws, size_t ws_size,
                              hipStream_t stream) {
    const float* un  = (const float*)d_in[0];   // unaries (1,H,W,C)
    const int*   spm = (const int*)  d_in[2];   // sp_map  (1,H,W)
    const float* sw  = (const float*)d_in[3];   // spatial_w (6,6)
    const float* bw  = (const float*)d_in[4];   // bilateral_w (6,6)
    const float* cm  = (const float*)d_in[5];   // compat (6,6)
    const float* lw  = (const float*)d_in[6];   // low_w (2,6)
    const float* hw  = (const float*)d_in[7];   // high_w (2,)
    float* qout = (float*)d_out;                // q in (p, c) layout == output layout

    // workspace layout (floats): [0,36) Mtot | [64,64+3072) Bsp | [3136,3136+3072) Bcont | segt ints
    float* ws   = (float*)d_ws;
    float* Mt   = ws;
    float* Bsp  = ws + 64;
    float* Bco  = ws + 64 + 3072;
    int*   segt = (int*)(ws + 64 + 6144);

    k_seg_transpose<<<dim3(WW / 32, HH / 32), dim3(32, 32), 0, stream>>>(spm, segt);
    k_mtot<<<1, 64, 0, stream>>>(cm, sw, bw, Mt);

    for (int it = 0; it < NITER; ++it) {
        const float* qsrc = (it == 0) ? un : qout;
        k_zero<<<24, 256, 0, stream>>>(Bsp, 6144);                       // zeros Bsp+Bco (contiguous)
        k_accum<<<PP / PIX_PER_BLK, 256, 0, stream>>>(qsrc, segt, Bsp, Bco);
        k_update<<<PP / 256, 256, 0, stream>>>(qsrc, un, segt, Bsp, Bco, Mt, lw, hw, qout);
    }
}